// lstm_crf_65541200937293
// MI455X (gfx1250) — compile-verified
//
#include <hip/hip_runtime.h>
#include <hip/hip_bf16.h>

// ---------------------------------------------------------------------------
// BiLSTM-CRF forward NLL for MI455X (gfx1250, wave32).
//
//  1) pack all weights into WMMA-B-operand order [tile][kstep][lane]{2f},
//     zero-padded -> branch-free, coalesced inner loops
//  2) embed gather -> X0 ([t][b] row order, width 52, zero padded)
//  3) hoisted input GEMMs  G = X @ Wih^T + b  via V_WMMA_F32_16X16X4_F32
//  4) persistent recurrent kernel: 16 WGs (2 dirs x 8 batch tiles of 16),
//     h/c/gates in LDS, packed Whh streamed from L2, fully unrolled 38-step
//     WMMA K-loop with unconditional loads
//  5) projection GEMM (unmasked; mask folded into CRF)
//  6) CRF gold + forward logsumexp scan, one WG per batch element
// ---------------------------------------------------------------------------

typedef __attribute__((ext_vector_type(2))) float v2f;
typedef __attribute__((ext_vector_type(8))) float v8f;

#define BATCH 128
#define TLEN  512
#define BT    (BATCH * TLEN)
#define EMB   50
#define EMBP  52     // padded to multiple of 4
#define HDIM  150
#define HPAD  152    // padded to multiple of 4
#define GDIM  600    // 4*H gates
#define GPAD  608    // padded to multiple of 16
#define GT    (GPAD / 16)   // 38 gate tiles
#define KS_H  (HPAD / 4)    // 38 k-steps for recurrent GEMM
#define HCAT  300
#define HCATP 304
#define KTAGS 48
#define SOS_TAG 2
#define NEGV  (-10000.0f)

// ---------------------------------------------------------------------------
// Pack W (N x ldw, row-major) into WMMA-B layout: out[tile][ks][lane]{2},
// where lane<16 covers rows k=ks*4..+1, lanes 16-31 rows k=ks*4+2..+3,
// column n = tile*16 + (lane&15).  OOB entries are zero.
__global__ __launch_bounds__(256)
void pack_weights_kernel(const float* __restrict__ W, int ldw, int N, int K,
                         int ksteps, int ntiles, float* __restrict__ out)
{
    int idx = blockIdx.x * blockDim.x + threadIdx.x;      // [tile][ks][lane]
    int total = ntiles * ksteps * 32;
    if (idx >= total) return;
    int lane = idx & 31;
    int rest = idx >> 5;
    int ks   = rest % ksteps;
    int tile = rest / ksteps;
    int n  = tile * 16 + (lane & 15);
    int kk = ks * 4 + 2 * (lane >> 4);
    float a = 0.0f, b = 0.0f;
    if (n < N) {
        if (kk     < K) a = W[(size_t)n * ldw + kk];
        if (kk + 1 < K) b = W[(size_t)n * ldw + kk + 1];
    }
    out[(size_t)idx * 2]     = a;
    out[(size_t)idx * 2 + 1] = b;
}

// ---------------------------------------------------------------------------
// Embedding gather into [t][b] row order, zero-padded rows of width EMBP.
__global__ __launch_bounds__(256)
void embed_kernel(const int* __restrict__ x, const float* __restrict__ embed,
                  float* __restrict__ X0, int total)
{
    int idx = blockIdx.x * blockDim.x + threadIdx.x;
    if (idx >= total) return;
    int r = idx / EMBP, c = idx - r * EMBP;     // r = t*BATCH + b
    float v = 0.0f;
    if (c < EMB) {
        int t = r / BATCH, b = r - t * BATCH;
        int tok = x[b * TLEN + t];
        v = embed[(size_t)tok * EMB + c];
    }
    X0[idx] = v;
}

__global__ __launch_bounds__(256)
void fill_zero_kernel(float* __restrict__ p, size_t n)
{
    size_t idx = (size_t)blockIdx.x * blockDim.x + threadIdx.x;
    if (idx < n) p[idx] = 0.0f;
}

// ---------------------------------------------------------------------------
// WMMA-f32 GEMM with pre-packed weights:
//   Out[m, n0:n0+16] = A[m,:] @ W^T + bias
// A (M x lda) zero-padded past K; Wpk packed as above.  Inner loop is
// branch-free: one LDS-free v2f A load + one coalesced v2f B load per WMMA.
__global__ __launch_bounds__(256)
void wmma_gemm_packed(const float* __restrict__ A, int lda,
                      const float* __restrict__ Wpk, int ksteps,
                      const float* __restrict__ bias,
                      float* __restrict__ Out, int ldo, int N)
{
    const int wave = threadIdx.x >> 5;
    const int lane = threadIdx.x & 31;
    const int half = lane >> 4;
    const int l16  = lane & 15;
    const int mbase = (blockIdx.x * 8 + wave) * 16;
    const int col   = blockIdx.y * 16 + l16;

    v8f acc;
    {
        const float bv = (col < N) ? bias[col] : 0.0f;
#pragma unroll
        for (int r = 0; r < 8; ++r) acc[r] = bv;
    }

    const float* __restrict__ arow = A + (size_t)(mbase + l16) * lda + 2 * half;
    const v2f*   __restrict__ wp   =
        (const v2f*)Wpk + ((size_t)blockIdx.y * ksteps) * 32 + lane;

    for (int ks = 0; ks < ksteps; ++ks) {
        v2f a = *(const v2f*)(arow + ks * 4);
        v2f b = wp[(size_t)ks * 32];
        acc = __builtin_amdgcn_wmma_f32_16x16x4_f32(
            false, a, false, b, (short)0, acc, false, false);
    }

#pragma unroll
    for (int r = 0; r < 8; ++r)
        Out[(size_t)(mbase + half * 8 + r) * ldo + col] = acc[r];
}

// ---------------------------------------------------------------------------
// Persistent bidirectional LSTM layer.  grid = 16 blocks:
//   blockIdx.x>>3 = direction (0 fwd, 1 bwd), blockIdx.x&7 = batch tile.
// 16 waves; each wave owns 2-3 of the 38 gate tiles.  h/c/gates in LDS.
// Inner K-loop fully unrolled: 38 x (LDS v2f A + coalesced packed-B v2f +
// WMMA f32), no guards.  Precomputed gates G ([t][b][GPAD]) are added in the
// coalesced elementwise phase.
__global__ __launch_bounds__(512)
void lstm_layer_kernel(const float* __restrict__ Gf,
                       const float* __restrict__ Gb,
                       const float* __restrict__ Wpkf,
                       const float* __restrict__ Wpkb,
                       const int*   __restrict__ x,
                       float* __restrict__ Hout /* [t*B+b] x HCATP */)
{
    __shared__ float sh_h[16][HPAD];   // pad cols 150,151 stay zero
    __shared__ float sh_c[16][HPAD];
    __shared__ float sh_g[16][GPAD];

    const int dir   = blockIdx.x >> 3;
    const int bbase = (blockIdx.x & 7) * 16;
    const float* __restrict__ G   = dir ? Gb   : Gf;
    const float* __restrict__ Wpk = dir ? Wpkb : Wpkf;

    for (int i = threadIdx.x; i < 16 * HPAD; i += blockDim.x) {
        (&sh_h[0][0])[i] = 0.0f;
        (&sh_c[0][0])[i] = 0.0f;
    }
    __syncthreads();

    const int wave = threadIdx.x >> 5;
    const int lane = threadIdx.x & 31;
    const int half = lane >> 4;
    const int l16  = lane & 15;

    for (int tt = 0; tt < TLEN; ++tt) {
        const int t = dir ? (TLEN - 1 - tt) : tt;

        // ---- phase A: sh_g(16 x 600) = h(16x150) @ Whh^T(150x600)
        for (int tile = wave; tile < GT; tile += 16) {
            const int col = tile * 16 + l16;
            const v2f* __restrict__ wp =
                (const v2f*)Wpk + ((size_t)tile * KS_H) * 32 + lane;
            v8f acc = {0.f, 0.f, 0.f, 0.f, 0.f, 0.f, 0.f, 0.f};
#pragma unroll
            for (int ks = 0; ks < KS_H; ++ks) {         // 38 WMMA ops
                v2f a = *(const v2f*)&sh_h[l16][ks * 4 + 2 * half];
                v2f b = wp[(size_t)ks * 32];
                acc = __builtin_amdgcn_wmma_f32_16x16x4_f32(
                    false, a, false, b, (short)0, acc, false, false);
            }
#pragma unroll
            for (int r = 0; r < 8; ++r)
                sh_g[half * 8 + r][col] = acc[r];
        }
        __syncthreads();

        // ---- phase B: add precomputed gates, activations, mask, carry
        for (int idx = threadIdx.x; idx < 16 * HDIM; idx += blockDim.x) {
            const int m = idx / HDIM;
            const int j = idx - m * HDIM;
            const int row = t * BATCH + bbase + m;
            const float* __restrict__ g = G + (size_t)row * GPAD;
            const float ig = sh_g[m][j]            + g[j];
            const float fg = sh_g[m][HDIM + j]     + g[HDIM + j];
            const float gg = sh_g[m][2 * HDIM + j] + g[2 * HDIM + j];
            const float og = sh_g[m][3 * HDIM + j] + g[3 * HDIM + j];
            const float is = 1.0f / (1.0f + __expf(-ig));
            const float fs = 1.0f / (1.0f + __expf(-fg));
            const float gt = tanhf(gg);
            const float os = 1.0f / (1.0f + __expf(-og));
            const float cn = fs * sh_c[m][j] + is * gt;
            const float hn = os * tanhf(cn);
            const bool msk = x[(bbase + m) * TLEN + t] > 0;
            Hout[(size_t)row * HCATP + dir * HDIM + j] = msk ? hn : 0.0f;
            sh_h[m][j] = msk ? hn : sh_h[m][j];
            sh_c[m][j] = msk ? cn : sh_c[m][j];
        }
        __syncthreads();
    }
}

// ---------------------------------------------------------------------------
// CRF: gold path score + forward partition, loss = Z - gold.
// Y is UNMASKED logits in [t][b] order; mask folded in here via x>0 guards
// (unmasked logits are never otherwise observable in the reference).
__global__ __launch_bounds__(64)
void crf_kernel(const float* __restrict__ Y,      // (T*B x 48)
                const int*   __restrict__ x,
                const int*   __restrict__ y0,
                const float* __restrict__ trans,  // (48 x 48) [next][prev]
                float* __restrict__ out)
{
    __shared__ float tr[KTAGS][KTAGS];
    __shared__ float s[KTAGS];
    __shared__ float snew[KTAGS];
    __shared__ float red[64];

    const int b   = blockIdx.x;
    const int tid = threadIdx.x;

    for (int i = tid; i < KTAGS * KTAGS; i += 64)
        (&tr[0][0])[i] = trans[i];

    // gold score (parallel over t, reduce); emit+transition only where mask=1
    float part = 0.0f;
    for (int t = tid; t < TLEN; t += 64) {
        const int xi = b * TLEN + t;          // token/tag index ([b][t])
        if (x[xi] > 0) {
            const int cur  = y0[xi];
            const int prev = (t == 0) ? SOS_TAG : y0[xi - 1];
            part += Y[(size_t)(t * BATCH + b) * KTAGS + cur];
            part += trans[cur * KTAGS + prev];
        }
    }
    red[tid] = part;
    if (tid < KTAGS) s[tid] = (tid == SOS_TAG) ? 0.0f : NEGV;
    __syncthreads();
    for (int off = 32; off > 0; off >>= 1) {
        if (tid < off) red[tid] += red[tid + off];
        __syncthreads();
    }
    const float gold = red[0];

    // forward scan (st only applied where mask=1, matching reference)
    for (int t = 0; t < TLEN; ++t) {
        const bool msk = x[b * TLEN + t] > 0;
        if (tid < KTAGS) {
            float mx = -3.4e38f;
#pragma unroll 4
            for (int p = 0; p < KTAGS; ++p)
                mx = fmaxf(mx, s[p] + tr[tid][p]);
            float sum = 0.0f;
#pragma unroll 4
            for (int p = 0; p < KTAGS; ++p)
                sum += __expf(s[p] + tr[tid][p] - mx);
            const float st = mx + __logf(sum)
                           + Y[(size_t)(t * BATCH + b) * KTAGS + tid];
            snew[tid] = msk ? st : s[tid];
        }
        __syncthreads();
        if (tid < KTAGS) s[tid] = snew[tid];
        __syncthreads();
    }

    if (tid == 0) {
        float mx = -3.4e38f;
        for (int p = 0; p < KTAGS; ++p) mx = fmaxf(mx, s[p]);
        float sum = 0.0f;
        for (int p = 0; p < KTAGS; ++p) sum += __expf(s[p] - mx);
        out[b] = (mx + __logf(sum)) - gold;
    }
}

// ---------------------------------------------------------------------------
extern "C" void kernel_launch(void* const* d_in, const int* in_sizes, int n_in,
                              void* d_out, int out_size, void* d_ws, size_t ws_size,
                              hipStream_t stream)
{
    // setup_inputs() flattening order:
    const int*   x     = (const int*)  d_in[0];
    const int*   y0    = (const int*)  d_in[1];
    const float* embed = (const float*)d_in[2];
    const float* Wih00 = (const float*)d_in[3];
    const float* Whh00 = (const float*)d_in[4];
    const float* b00   = (const float*)d_in[5];
    const float* Wih01 = (const float*)d_in[6];
    const float* Whh01 = (const float*)d_in[7];
    const float* b01   = (const float*)d_in[8];
    const float* Wih10 = (const float*)d_in[9];
    const float* Whh10 = (const float*)d_in[10];
    const float* b10   = (const float*)d_in[11];
    const float* Wih11 = (const float*)d_in[12];
    const float* Whh11 = (const float*)d_in[13];
    const float* b11   = (const float*)d_in[14];
    const float* outW  = (const float*)d_in[15];
    const float* outb  = (const float*)d_in[16];
    const float* trans = (const float*)d_in[17];
    (void)ws_size; (void)in_sizes; (void)n_in; (void)out_size;

    char* w = (char*)d_ws;
    size_t off = 0;
    auto salloc = [&](size_t elems) -> float* {
        float* p = (float*)(w + off);
        off = (off + elems * 4 + 255) & ~(size_t)255;
        return p;
    };
    // activations
    float* X0  = salloc((size_t)BT * EMBP);
    float* Gx0 = salloc((size_t)BT * GPAD);
    float* Gx1 = salloc((size_t)BT * GPAD);
    float* H1  = salloc((size_t)BT * HCATP);
    float* H2  = salloc((size_t)BT * HCATP);
    float* Yb  = salloc((size_t)BT * KTAGS);
    // packed weights
    const int KS_E = EMBP / 4, KS_C = HCATP / 4;          // 13, 76
    float* pWih00 = salloc((size_t)GT * KS_E * 64);
    float* pWih01 = salloc((size_t)GT * KS_E * 64);
    float* pWih10 = salloc((size_t)GT * KS_C * 64);
    float* pWih11 = salloc((size_t)GT * KS_C * 64);
    float* pWhh00 = salloc((size_t)GT * KS_H * 64);
    float* pWhh01 = salloc((size_t)GT * KS_H * 64);
    float* pWhh10 = salloc((size_t)GT * KS_H * 64);
    float* pWhh11 = salloc((size_t)GT * KS_H * 64);
    float* pOutW  = salloc((size_t)3  * KS_C * 64);

    // 0) pack all weights (tiny, one pass each)
    auto pack = [&](const float* W, int ldw, int N, int K, int ksteps,
                    int ntiles, float* dst) {
        int total = ntiles * ksteps * 32;
        pack_weights_kernel<<<(total + 255) / 256, 256, 0, stream>>>(
            W, ldw, N, K, ksteps, ntiles, dst);
    };
    pack(Wih00, EMB,  GDIM, EMB,  KS_E, GT, pWih00);
    pack(Wih01, EMB,  GDIM, EMB,  KS_E, GT, pWih01);
    pack(Wih10, HCAT, GDIM, HCAT, KS_C, GT, pWih10);
    pack(Wih11, HCAT, GDIM, HCAT, KS_C, GT, pWih11);
    pack(Whh00, HDIM, GDIM, HDIM, KS_H, GT, pWhh00);
    pack(Whh01, HDIM, GDIM, HDIM, KS_H, GT, pWhh01);
    pack(Whh10, HDIM, GDIM, HDIM, KS_H, GT, pWhh10);
    pack(Whh11, HDIM, GDIM, HDIM, KS_H, GT, pWhh11);
    pack(outW,  HCAT, KTAGS, HCAT, KS_C, 3, pOutW);

    // 1) embedding gather ([t][b] order, zero pad cols)
    {
        int total = BT * EMBP;
        embed_kernel<<<(total + 255) / 256, 256, 0, stream>>>(x, embed, X0, total);
    }
    // zero-init hidden buffers (pad cols 300..303 must read as zero)
    {
        size_t n = (size_t)BT * HCATP;
        fill_zero_kernel<<<(int)((n + 255) / 256), 256, 0, stream>>>(H1, n);
        fill_zero_kernel<<<(int)((n + 255) / 256), 256, 0, stream>>>(H2, n);
    }

    dim3 gGates(BT / 128, GT);   // 512 x 38 tiles

    // 2) layer 0: input-gate precompute
    wmma_gemm_packed<<<gGates, 256, 0, stream>>>(X0, EMBP, pWih00, KS_E, b00,
                                                 Gx0, GPAD, GDIM);
    wmma_gemm_packed<<<gGates, 256, 0, stream>>>(X0, EMBP, pWih01, KS_E, b01,
                                                 Gx1, GPAD, GDIM);
    // 3) layer 0 recurrence
    lstm_layer_kernel<<<16, 512, 0, stream>>>(Gx0, Gx1, pWhh00, pWhh01, x, H1);

    // 4) layer 1: input-gate precompute
    wmma_gemm_packed<<<gGates, 256, 0, stream>>>(H1, HCATP, pWih10, KS_C, b10,
                                                 Gx0, GPAD, GDIM);
    wmma_gemm_packed<<<gGates, 256, 0, stream>>>(H1, HCATP, pWih11, KS_C, b11,
                                                 Gx1, GPAD, GDIM);
    // 5) layer 1 recurrence
    lstm_layer_kernel<<<16, 512, 0, stream>>>(Gx0, Gx1, pWhh10, pWhh11, x, H2);

    // 6) projection to tags (unmasked; CRF applies the mask)
    dim3 gProj(BT / 128, KTAGS / 16);
    wmma_gemm_packed<<<gProj, 256, 0, stream>>>(H2, HCATP, pOutW, KS_C, outb,
                                                Yb, KTAGS, KTAGS);

    // 7) CRF loss
    crf_kernel<<<BATCH, 64, 0, stream>>>(Yb, x, y0, trans, (float*)d_out);
}